// AttributePredictor_22952305230274
// MI455X (gfx1250) — compile-verified
//
#include <hip/hip_runtime.h>
#include <hip/hip_bf16.h>
#include <math.h>
#include <stdint.h>

// ---------------- problem constants (match reference) ----------------
#define BS     32
#define NT     196
#define DV     1024
#define NBOX   512
#define GRID_N 14
#define SCALE_F 0.0625f          // 14/224
#define MAX_ATT 9
#define LN_EPS 1e-5f

__constant__ int kID2CAT[20] = {3, 5, 2, 4, 6, 3, 7, 2, 5, 4,
                                8, 3, 2, 6, 5, 4, 3, 9, 2, 5};

typedef __attribute__((ext_vector_type(2))) float v2f;
typedef __attribute__((ext_vector_type(8))) float v8f;

// ---- CDNA5 async global->LDS copy (ASYNCcnt-tracked, no VGPR bounce) ----
// LDS operand is the wave-relative LDS byte offset == low 32 bits of the
// flat pointer into the LDS aperture.
__device__ __forceinline__ void async_copy_b128(const float* gsrc, void* ldst)
{
    const uint32_t loff = (uint32_t)(uintptr_t)ldst;
    asm volatile("global_load_async_to_lds_b128 %0, %1, off"
                 :
                 : "v"(loff), "v"(gsrc)
                 : "memory");
}
__device__ __forceinline__ void wait_asynccnt0()
{
    asm volatile("s_wait_asynccnt 0x0" ::: "memory");
}

// =====================================================================
// Kernel 1: ROI max pool (1x1 output). One block per box, 256 threads,
// 4 channels per thread (d = tid + i*256 -> fully coalesced per cell).
// feats[b,d,r,c] == x[b, r*14+c, d]
// =====================================================================
__global__ void __launch_bounds__(256)
roi_pool_kernel(const float* __restrict__ x,
                const float* __restrict__ boxes,
                float* __restrict__ pooled)
{
    const int n   = blockIdx.x;
    const int tid = threadIdx.x;

    const int bidx = (int)boxes[n * 5 + 0];
    const int x1 = (int)floorf(boxes[n * 5 + 1] * SCALE_F + 0.5f);
    const int y1 = (int)floorf(boxes[n * 5 + 2] * SCALE_F + 0.5f);
    const int x2 = (int)floorf(boxes[n * 5 + 3] * SCALE_F + 0.5f);
    const int y2 = (int)floorf(boxes[n * 5 + 4] * SCALE_F + 0.5f);

    int hs = min(max(y1, 0), GRID_N);
    int he = min(max(max(y2 + 1, y1 + 1), 0), GRID_N);
    int ws = min(max(x1, 0), GRID_N);
    int we = min(max(max(x2 + 1, x1 + 1), 0), GRID_N);
    const bool empty = (hs >= he) || (ws >= we);

    const float* xb = x + (size_t)bidx * NT * DV;

    float m0 = -INFINITY, m1 = -INFINITY, m2 = -INFINITY, m3 = -INFINITY;
    for (int r = hs; r < he; ++r) {
        for (int c = ws; c < we; ++c) {
            const float* p = xb + (size_t)(r * GRID_N + c) * DV;
            m0 = fmaxf(m0, p[tid + 0 * 256]);
            m1 = fmaxf(m1, p[tid + 1 * 256]);
            m2 = fmaxf(m2, p[tid + 2 * 256]);
            m3 = fmaxf(m3, p[tid + 3 * 256]);
        }
    }
    float* out = pooled + (size_t)n * DV;
    out[tid + 0 * 256] = empty ? 0.0f : m0;
    out[tid + 1 * 256] = empty ? 0.0f : m1;
    out[tid + 2 * 256] = empty ? 0.0f : m2;
    out[tid + 3 * 256] = empty ? 0.0f : m3;
}

// =====================================================================
// Kernel 2: h = pooled @ W_ff^T + b_ff  via V_WMMA_F32_16X16X4_F32.
//   M=512 (boxes), N=1024 (out features), K=1024.
//   Block = 256 threads (8 wave32), tile = 16(M) x 128(N); each wave
//   owns one 16x16 sub-tile and runs 256 WMMAs across K.
// LDS staged via GLOBAL_LOAD_ASYNC_TO_LDS_B128 (ASYNCcnt), pad stride 36
// floats (144 B): keeps every float4 LDS dest 16B-aligned AND fragment
// reads bank-conflict free (36*i mod 64 distinct for i=0..15).
// A frag (16x4 f32): lanes 0-15 hold K=0,1 ; lanes 16-31 hold K=2,3.
// B frag (4x16 f32): VGPR0 = rows K=0 (lanes 0-15) / K=2 (lanes 16-31),
//                    VGPR1 = rows K=1 / K=3  (mirrors C/D half-split).
// =====================================================================
#define BM 16
#define BN 128
#define KC 32
#define LDA 36
#define LDB 36

__global__ void __launch_bounds__(256)
ffn_gemm_kernel(const float* __restrict__ pooled,
                const float* __restrict__ Wff,
                const float* __restrict__ bff,
                float* __restrict__ h)
{
    __shared__ __align__(16) float As[BM * LDA];
    __shared__ __align__(16) float Bs[BN * LDB];

    const int tid  = threadIdx.x;
    const int lane = tid & 31;
    const int wave = tid >> 5;                 // 0..7
    const int m0   = blockIdx.x * BM;          // 0..511
    const int n0   = blockIdx.y * BN;          // 0..1023

    const int mrow  = lane & 15;
    const int khalf = (lane >> 4) << 1;        // 0 (lanes 0-15) or 2 (lanes 16-31)
    const int ncol  = wave * 16 + (lane & 15);

    // Fixed staging assignments (unrolled, float4 granularity):
    //  A tile 16x32 = 512 floats -> threads 0..127, one b128 each.
    const int eA  = tid * 4;
    const int amA = eA >> 5;
    const int akA = eA & 31;

    v8f acc = {};

    for (int k0 = 0; k0 < DV; k0 += KC) {
        // ---- async stage A tile: 16x32 floats ----
        if (tid < 128) {
            async_copy_b128(&pooled[(size_t)(m0 + amA) * DV + k0 + akA],
                            &As[amA * LDA + akA]);
        }
        // ---- async stage B tile: 128x32 floats, 4 x b128 per thread ----
        #pragma unroll
        for (int j = 0; j < 4; ++j) {
            const int e  = tid * 4 + j * 1024;
            const int nn = e >> 5;
            const int kk = e & 31;
            async_copy_b128(&Wff[(size_t)(n0 + nn) * DV + k0 + kk],
                            &Bs[nn * LDB + kk]);
            if (k0 + KC < DV)    // hint next K chunk toward the caches
                __builtin_prefetch(&Wff[(size_t)(n0 + nn) * DV + k0 + KC + kk], 0, 1);
        }
        wait_asynccnt0();
        __syncthreads();

        #pragma unroll
        for (int kk = 0; kk < KC; kk += 4) {
            v2f a, b;
            a[0] = As[mrow * LDA + kk + khalf + 0];
            a[1] = As[mrow * LDA + kk + khalf + 1];
            b[0] = Bs[ncol * LDB + kk + khalf + 0];
            b[1] = Bs[ncol * LDB + kk + khalf + 1];
            // 8 args: (neg_a, A, neg_b, B, c_mod, C, reuse_a, reuse_b)
            acc = __builtin_amdgcn_wmma_f32_16x16x4_f32(
                false, a, false, b, (short)0, acc, false, false);
        }
        __syncthreads();
    }

    // ---- writeback: C/D layout -> lanes 0-15: M=v, lanes 16-31: M=v+8 ----
    const int col   = n0 + wave * 16 + (lane & 15);
    const int rbase = m0 + ((lane >> 4) << 3);
    const float bias = bff[col];
    #pragma unroll
    for (int v = 0; v < 8; ++v) {
        h[(size_t)(rbase + v) * DV + col] = acc[v] + bias;
    }
}

// =====================================================================
// Kernel 3: LayerNorm + exact GELU + attribute/disr heads.
// One block (256 thr) per box; y kept in LDS for the 10 dot products.
// =====================================================================
__device__ __forceinline__ float block_reduce_sum(float v, float* red, int tid)
{
    red[tid] = v;
    __syncthreads();
    #pragma unroll
    for (int s = 128; s > 0; s >>= 1) {
        if (tid < s) red[tid] += red[tid + s];
        __syncthreads();
    }
    const float r = red[0];
    __syncthreads();
    return r;
}

__global__ void __launch_bounds__(256)
ln_gelu_heads_kernel(const float* __restrict__ h,
                     const float* __restrict__ ln_g,
                     const float* __restrict__ ln_b,
                     const int*   __restrict__ labels,
                     const float* __restrict__ Wh,
                     const float* __restrict__ bh,
                     const float* __restrict__ Wd,
                     const float* __restrict__ bd,
                     float* __restrict__ y_out,
                     float* __restrict__ logits_out,
                     float* __restrict__ disr_out)
{
    __shared__ float ysh[DV];
    __shared__ float red[256];

    const int n   = blockIdx.x;
    const int tid = threadIdx.x;
    const float* hr = h + (size_t)n * DV;

    float hv[4];
    float s = 0.0f, ss = 0.0f;
    #pragma unroll
    for (int i = 0; i < 4; ++i) {
        hv[i] = hr[tid + i * 256];
        s  += hv[i];
        ss += hv[i] * hv[i];
    }
    const float sum  = block_reduce_sum(s,  red, tid);
    const float sum2 = block_reduce_sum(ss, red, tid);
    const float mu   = sum * (1.0f / DV);
    const float var  = sum2 * (1.0f / DV) - mu * mu;
    const float rstd = rsqrtf(var + LN_EPS);

    #pragma unroll
    for (int i = 0; i < 4; ++i) {
        const int d  = tid + i * 256;
        const float t  = (hv[i] - mu) * rstd * ln_g[d] + ln_b[d];
        const float yv = 0.5f * t * (1.0f + erff(t * 0.70710678118654752f));
        ysh[d] = yv;
        y_out[(size_t)n * DV + d] = yv;
    }
    __syncthreads();

    const int label = labels[n];
    const int nact  = kID2CAT[label];

    // 9 attribute logits (masked) + 1 disr logit
    for (int k = 0; k < MAX_ATT; ++k) {
        const float* w = Wh + ((size_t)label * MAX_ATT + k) * DV;
        float p = 0.0f;
        #pragma unroll
        for (int i = 0; i < 4; ++i) {
            const int d = tid + i * 256;
            p += ysh[d] * w[d];
        }
        const float tot = block_reduce_sum(p, red, tid);
        if (tid == 0) {
            logits_out[n * MAX_ATT + k] =
                (k < nact) ? (tot + bh[label * MAX_ATT + k]) : 0.0f;  // ATT_PAD
        }
    }
    {
        float p = 0.0f;
        #pragma unroll
        for (int i = 0; i < 4; ++i) {
            const int d = tid + i * 256;
            p += ysh[d] * Wd[d];
        }
        const float tot = block_reduce_sum(p, red, tid);
        if (tid == 0) disr_out[n] = tot + bd[0];
    }
}

// =====================================================================
// Host-side launcher
// inputs: x, boxes, box_labels, W_ff, b_ff, ln_g, ln_b, Wh, bh, Wd, bd
// output: y (512*1024) | logits (512*9) | disr (512*1)   (fp32)
// =====================================================================
extern "C" void kernel_launch(void* const* d_in, const int* in_sizes, int n_in,
                              void* d_out, int out_size, void* d_ws, size_t ws_size,
                              hipStream_t stream)
{
    (void)in_sizes; (void)n_in; (void)out_size; (void)ws_size;

    const float* x      = (const float*)d_in[0];
    const float* boxes  = (const float*)d_in[1];
    const int*   labels = (const int*)  d_in[2];
    const float* W_ff   = (const float*)d_in[3];
    const float* b_ff   = (const float*)d_in[4];
    const float* ln_g   = (const float*)d_in[5];
    const float* ln_b   = (const float*)d_in[6];
    const float* Wh     = (const float*)d_in[7];
    const float* bh     = (const float*)d_in[8];
    const float* Wd     = (const float*)d_in[9];
    const float* bd     = (const float*)d_in[10];

    float* pooled = (float*)d_ws;                       // 512*1024 floats
    float* hbuf   = pooled + (size_t)NBOX * DV;         // 512*1024 floats

    float* y_out      = (float*)d_out;
    float* logits_out = y_out + (size_t)NBOX * DV;
    float* disr_out   = logits_out + (size_t)NBOX * MAX_ATT;

    roi_pool_kernel<<<NBOX, 256, 0, stream>>>(x, boxes, pooled);

    dim3 gemm_grid(NBOX / BM, DV / BN);                 // 32 x 8
    ffn_gemm_kernel<<<gemm_grid, 256, 0, stream>>>(pooled, W_ff, b_ff, hbuf);

    ln_gelu_heads_kernel<<<NBOX, 256, 0, stream>>>(
        hbuf, ln_g, ln_b, labels, Wh, bh, Wd, bd,
        y_out, logits_out, disr_out);
}